// gpt2xcnn_32650341384218
// MI455X (gfx1250) — compile-verified
//
#include <hip/hip_runtime.h>
#include <math.h>

// ---------------- problem constants (from reference) ----------------
#define B_BATCH 32
#define N_ACT   1024
#define N_FULL  2048
#define T_LEN   2048
#define NF      113      // frames
#define NBINS   40
#define FRAME   256
#define STEP    16
#define NCOLS   80       // 2*NBINS, interleaved Re/Im columns

typedef __attribute__((ext_vector_type(2))) float v2f;
typedef __attribute__((ext_vector_type(4))) float v4f;
typedef __attribute__((ext_vector_type(8))) float v8f;

// ---------------- workspace layout (float offsets) ----------------
// Dt[80][256] | acc[32] | sig_part[2][32][3][2048]  (two K-half partials)
#define WS_DT   0
#define WS_ACC  (NCOLS * FRAME)
#define WS_SIG  (WS_ACC + 32)

// =====================================================================
// Kernel 0: build windowed DFT basis Dt[col][n], col even -> w*cos,
// col odd -> -w*sin, for bin k=col/2. Also zero the batch accumulators.
// Phase reduced mod 256 in integers so sinf/cosf stay accurate.
// =====================================================================
__global__ void build_dft_kernel(float* __restrict__ Dt, float* __restrict__ acc) {
    const int col = blockIdx.x;      // 0..79
    const int n   = threadIdx.x;     // 0..255
    const float TWO_PI = 6.28318530717958647692f;
    const float w = 0.5f - 0.5f * cosf(TWO_PI * (float)n * (1.0f / 256.0f));
    const int   k = col >> 1;
    const int   phase = (k * n) & 255;                 // exact mod-256 reduction
    const float th = TWO_PI * (float)phase * (1.0f / 256.0f);
    const float val = (col & 1) ? (-w * sinf(th)) : (w * cosf(th));
    Dt[col * FRAME + n] = val;
    if (blockIdx.x == 0 && n < B_BATCH) acc[n] = 0.0f;
}

// =====================================================================
// Kernel 1: fused scatter + projection (HBM-bound streaming reduction)
// signals[b,c,t] = sum_a tmat[11+c, idx[a]] * sigs[b,a,t]
// Split over K halves (512 rows each) for grid parallelism; partials
// are summed during the STFT kernel's LDS staging.
// Grid: (4 t-chunks of 512, 2 k-halves, B). Block: 128 thr x float4.
// b128 non-temporal loads: 512 B / wave / instruction, streamed once.
// =====================================================================
__global__ void project_kernel(const float* __restrict__ sigs,
                               const float* __restrict__ tmat,
                               const int*   __restrict__ lidx,
                               const int*   __restrict__ ridx,
                               float*       __restrict__ sig_part) {
    __shared__ float Aw[3 * 512];                // this K-half's weights, 6 KB
    const int tid = threadIdx.x;                 // 0..127
    const int kh  = blockIdx.y;                  // 0..1
    const int b   = blockIdx.z;
    const int t0  = blockIdx.x * 512 + tid * 4;

    for (int i = tid; i < 3 * 512; i += 128) {
        const int c = i >> 9;                    // /512
        const int al = i & 511;
        const int ag = kh * 512 + al;            // global active-vertex index
        const int col = (ag < 512) ? lidx[ag] : ridx[ag - 512];
        Aw[i] = tmat[(11 + c) * N_FULL + col];
    }
    __syncthreads();

    v4f a0 = {0.f,0.f,0.f,0.f}, a1 = a0, a2 = a0;
    const float* srow = sigs + (size_t)b * N_ACT * T_LEN
                             + (size_t)kh * 512 * T_LEN + t0;

    #pragma unroll 8
    for (int a = 0; a < 512; ++a) {
        // streamed exactly once -> non-temporal b128 load
        const v4f s = __builtin_nontemporal_load((const v4f*)(srow + (size_t)a * T_LEN));
        const float w0 = Aw[a];
        const float w1 = Aw[512 + a];
        const float w2 = Aw[1024 + a];
        a0.x = fmaf(w0, s.x, a0.x); a0.y = fmaf(w0, s.y, a0.y);
        a0.z = fmaf(w0, s.z, a0.z); a0.w = fmaf(w0, s.w, a0.w);
        a1.x = fmaf(w1, s.x, a1.x); a1.y = fmaf(w1, s.y, a1.y);
        a1.z = fmaf(w1, s.z, a1.z); a1.w = fmaf(w1, s.w, a1.w);
        a2.x = fmaf(w2, s.x, a2.x); a2.y = fmaf(w2, s.y, a2.y);
        a2.z = fmaf(w2, s.z, a2.z); a2.w = fmaf(w2, s.w, a2.w);
    }

    float* out = sig_part + ((size_t)(kh * B_BATCH + b) * 3) * T_LEN + t0;
    *(v4f*)(out)             = a0;
    *(v4f*)(out + T_LEN)     = a1;
    *(v4f*)(out + 2 * T_LEN) = a2;
}

// =====================================================================
// Kernel 2: STFT-as-GEMM via V_WMMA_F32_16X16X4_F32, fused magnitude +
// classifier dot-product. Grid: (3 channels, B). Block: 256 = 8 waves;
// wave w owns frame-tile mtile=w (16 frames). K=256 in 64 steps of 4.
// Staging loop also sums the two projection K-half partials.
// A fragment: 16x4 frame window slice from LDS (frames overlap, step 16).
// B fragment: Dt[col][k] pairs, 8B global loads, L2-resident.
// C layout: vgpr r, lanes 0-15 -> M=r, lanes 16-31 -> M=r+8; N=lane%16.
// =====================================================================
__global__ void stft_wmma_kernel(const float* __restrict__ sig_part,
                                 const float* __restrict__ Dt,
                                 const float* __restrict__ Wc,
                                 float*       __restrict__ acc) {
    __shared__ float srow[2304];   // 2048 signal + zero tail (frame 127 reads to 2287)
    const int tid = threadIdx.x;
    const int c   = blockIdx.x;    // channel 0..2
    const int b   = blockIdx.y;    // batch

    const float* row0 = sig_part + ((size_t)(b * 3 + c)) * T_LEN;
    const float* row1 = sig_part + ((size_t)((B_BATCH + b) * 3 + c)) * T_LEN;
    for (int i = tid; i < T_LEN; i += 256) srow[i] = row0[i] + row1[i];
    srow[T_LEN + tid] = 0.0f;      // zero-pad 2048..2303
    __syncthreads();

    const int mtile = tid >> 5;          // 0..7  (frame tile)
    const int lane  = tid & 31;
    const int n16   = lane & 15;
    const int koff  = (lane < 16) ? 0 : 2;   // A/B K split across lane halves
    const int f_a   = mtile * 16 + n16;      // frame row this lane feeds into A

    v8f cacc[5];
    #pragma unroll
    for (int nt = 0; nt < 5; ++nt) cacc[nt] = (v8f){0.f,0.f,0.f,0.f,0.f,0.f,0.f,0.f};

    const float* dtl = Dt + (size_t)n16 * FRAME + koff;   // + nt*16*FRAME per N tile

    for (int ks = 0; ks < 64; ++ks) {
        const int kbase = ks * 4;
        // A: frames overlap -> element (f*STEP + k) of the staged row
        const v2f afrag = *(const v2f*)&srow[f_a * STEP + kbase + koff];
        #pragma unroll
        for (int nt = 0; nt < 5; ++nt) {
            const v2f bfrag = *(const v2f*)(dtl + (size_t)nt * 16 * FRAME + kbase);
            cacc[nt] = __builtin_amdgcn_wmma_f32_16x16x4_f32(
                false, afrag, false, bfrag, (short)0, cacc[nt], false, false);
        }
    }

    // magnitude (Re/Im are adjacent columns -> adjacent lanes) + W_cls dot
    float partial = 0.0f;
    const int mbase = mtile * 16 + ((lane >= 16) ? 8 : 0);
    #pragma unroll
    for (int nt = 0; nt < 5; ++nt) {
        const int  n    = nt * 16 + n16;
        const int  kbin = n >> 1;
        const bool even = (n & 1) == 0;
        #pragma unroll
        for (int r = 0; r < 8; ++r) {
            const float v = cacc[nt][r];
            const float o = __shfl_xor(v, 1, 32);       // Re<->Im partner
            const int   f = mbase + r;
            if (even && f < NF) {
                const float mag = sqrtf(v * v + o * o);
                partial += mag * Wc[(c * NF + f) * NBINS + kbin];
            }
        }
    }

    #pragma unroll
    for (int off = 16; off > 0; off >>= 1)
        partial += __shfl_down(partial, off, 32);
    if (lane == 0) atomicAdd(&acc[b], partial);
}

// =====================================================================
// Kernel 3: bias + sigmoid
// =====================================================================
__global__ void finalize_kernel(const float* __restrict__ acc,
                                const float* __restrict__ bcls,
                                float*       __restrict__ out) {
    const int i = threadIdx.x;
    if (i < B_BATCH) {
        const float z = acc[i] + bcls[0];
        out[i] = 1.0f / (1.0f + expf(-z));
    }
}

// =====================================================================
extern "C" void kernel_launch(void* const* d_in, const int* in_sizes, int n_in,
                              void* d_out, int out_size, void* d_ws, size_t ws_size,
                              hipStream_t stream) {
    const float* sigs = (const float*)d_in[0];
    const float* tmat = (const float*)d_in[1];
    const int*   lidx = (const int*)d_in[2];
    const int*   ridx = (const int*)d_in[3];
    const float* Wc   = (const float*)d_in[4];
    const float* bcls = (const float*)d_in[5];

    float* ws       = (float*)d_ws;
    float* Dt       = ws + WS_DT;
    float* acc      = ws + WS_ACC;
    float* sig_part = ws + WS_SIG;

    hipLaunchKernelGGL(build_dft_kernel, dim3(NCOLS), dim3(FRAME), 0, stream, Dt, acc);
    hipLaunchKernelGGL(project_kernel,   dim3(4, 2, B_BATCH), dim3(128), 0, stream,
                       sigs, tmat, lidx, ridx, sig_part);
    hipLaunchKernelGGL(stft_wmma_kernel, dim3(3, B_BATCH), dim3(256), 0, stream,
                       sig_part, Dt, Wc, acc);
    hipLaunchKernelGGL(finalize_kernel,  dim3(1), dim3(32), 0, stream,
                       acc, bcls, (float*)d_out);
}